// Position_linear_58256936403651
// MI455X (gfx1250) — compile-verified
//
#include <hip/hip_runtime.h>
#include <stdint.h>

#define SEQ     256
#define FEATURE 768
#define WINDOW  3
#define FILTER  6
#define KTOT    (WINDOW * FEATURE)     // 2304
#define WFLOATS (KTOT * FILTER)        // 13824 floats = 55.3 KB
#define WVEC4   (WFLOATS / 4)          // 3456 16-byte chunks

typedef float v2f __attribute__((ext_vector_type(2)));
typedef float v8f __attribute__((ext_vector_type(8)));

// One block per position s: 512 threads = 16 waves, wave w owns batch M-tile w.
// W[s] (2304x6 f32) is staged into LDS with gfx1250 async-to-LDS B128 copies
// (ASYNCcnt path, no VGPR round trip), then each wave runs a K=2304 f32 WMMA
// chain (V_WMMA_F32_16X16X4_F32) with 4 independent accumulators.
__global__ __launch_bounds__(512) void poslin_wmma_f32(
    const float* __restrict__ X,     // (256, SEQ, FEATURE)
    const float* __restrict__ Wm,    // (SEQ, KTOT, FILTER)
    const float* __restrict__ Bias,  // (SEQ, FILTER)
    float* __restrict__ Out)         // (256, SEQ, FILTER)
{
    const int s    = blockIdx.x;
    const int tid  = threadIdx.x;
    const int wave = tid >> 5;      // 0..15 -> M-tile
    const int lane = tid & 31;
    const int half = lane >> 4;     // 0: K pair {0,1}, 1: K pair {2,3}
    const int rowc = lane & 15;     // A row / B-C-D column
    const int kb   = half * 2;

    __shared__ float ldsB[WFLOATS];

    // ---- Async-stage W[s] into LDS: global_load_async_to_lds_b128 ----
    {
        // Low 32 bits of a generic LDS pointer == allocation-relative LDS
        // byte offset (same space ds_* / async-LDS VDST expects).
        const unsigned ldsBase = (unsigned)(uintptr_t)(void*)&ldsB[0];
        const uint64_t gbase   = (uint64_t)(uintptr_t)(Wm + (size_t)s * WFLOATS);
        for (int i = tid; i < WVEC4; i += 512) {
            unsigned laddr = ldsBase + (unsigned)i * 16u;
            uint64_t gaddr = gbase + (uint64_t)i * 16u;
            asm volatile("global_load_async_to_lds_b128 %0, %1, off"
                         :: "v"(laddr), "v"(gaddr)
                         : "memory");
        }
        asm volatile("s_wait_asynccnt 0x0" ::: "memory");
    }
    __syncthreads();

    const int bm = wave * 16;                    // first batch row of tile
    // B column for this lane; cols >= 6 are padding -> clamp (those output
    // columns are never stored, garbage is fine)
    const int cc = (rowc < FILTER) ? rowc : (FILTER - 1);

    // 4 independent accumulators to break the WMMA->WMMA RAW chain
    v8f acc0 = {}, acc1 = {}, acc2 = {}, acc3 = {};

#define WMMA_STEP(ACC, KK)                                                    \
    do {                                                                      \
        v2f a = *(const v2f*)(aptr + (KK));                                   \
        v2f b;                                                                \
        b.x = bbase[((KK) + kb)     * FILTER];                                \
        b.y = bbase[((KK) + kb + 1) * FILTER];                                \
        ACC = __builtin_amdgcn_wmma_f32_16x16x4_f32(                          \
            false, a, false, b, (short)0, ACC, false, false);                 \
    } while (0)

    for (int w = 0; w < WINDOW; ++w) {
        const int t = s + w - 1;                 // source timestep
        if (t < 0 || t >= SEQ) continue;         // zero-padded window: skip == +0
        const float* aptr  = X + ((size_t)(bm + rowc) * SEQ + t) * FEATURE + kb;
        const float* bbase = ldsB + (w * FEATURE) * FILTER + cc;

        for (int k = 0; k < FEATURE; k += 16) {
            WMMA_STEP(acc0, k);
            WMMA_STEP(acc1, k + 4);
            WMMA_STEP(acc2, k + 8);
            WMMA_STEP(acc3, k + 12);
        }
    }
#undef WMMA_STEP

    v8f acc = (acc0 + acc1) + (acc2 + acc3);

    // ---- Epilogue: bias + ReLU + store the 6 real columns ----
    if (rowc < FILTER) {
        const float bb = Bias[s * FILTER + rowc];
#pragma unroll
        for (int v = 0; v < 8; ++v) {
            const int m = v + 8 * half;          // C/D layout: VGPR v, lane half
            float r = acc[v] + bb;
            r = fmaxf(r, 0.0f);
            Out[((size_t)(bm + m) * SEQ + s) * FILTER + rowc] = r;
        }
    }
}

extern "C" void kernel_launch(void* const* d_in, const int* in_sizes, int n_in,
                              void* d_out, int out_size, void* d_ws, size_t ws_size,
                              hipStream_t stream)
{
    (void)in_sizes; (void)n_in; (void)d_ws; (void)ws_size; (void)out_size;
    const float* X    = (const float*)d_in[0];  // inputs (256, 256, 768) f32
    const float* Wm   = (const float*)d_in[1];  // W      (256, 2304, 6)  f32
    const float* Bias = (const float*)d_in[2];  // b      (256, 6)        f32
    float* Out        = (float*)d_out;          // out    (256, 256, 6)   f32

    poslin_wmma_f32<<<dim3(SEQ), 512, 0, stream>>>(X, Wm, Bias, Out);
}